// SortModel_73289321939595
// MI455X (gfx1250) — compile-verified
//
#include <hip/hip_runtime.h>
#include <hip/hip_bf16.h>

typedef unsigned int u32;
typedef unsigned long long u64;

#define BLK     256
#define KPT     16               // items per thread
#define TILE    (BLK * KPT)      // 4096 elements per block
#define RPASSES 4                // 4 x 8-bit digits
#define RED_BLOCKS 4096

typedef __attribute__((ext_vector_type(2))) float v2f;
typedef __attribute__((ext_vector_type(8))) float v8f;
typedef __attribute__((ext_vector_type(4))) unsigned int v4u;
typedef __attribute__((ext_vector_type(8))) int v8i;
typedef __attribute__((ext_vector_type(4))) int v4i;

// ---------------------------------------------------------------------------
// Tensor Data Mover: 1-row 2D descriptor, contiguous tile -> LDS.
// D# packing per CDNA5 ISA:
//   g0: [1:0]=count=1, [63:32]=lds_addr, [120:64]=global_addr, [127:126]=type=2
//   g1: data_size=2 (4B), tensor_dim0=remaining len (OOB reads -> 0),
//       tensor_dim1=1, tile_dim0=tile_len, tile_dim1=1, dim0_stride=len.
// Trailing groups zero-filled (only needed for >2D tensors).
// LDS offset: low 32 bits of the generic address of a __shared__ object are
// the wave-relative LDS byte offset (flat-LDS aperture truncation rule).
// This toolchain's builtin is the 6-arg form:
//   (uint32x4 g0, int32x8 g1, int32x4, int32x4, int32x8, i32 cpol)
// ---------------------------------------------------------------------------
__device__ __forceinline__ unsigned lds_off(const void* p) {
  return (unsigned)(uintptr_t)p;
}

__device__ __forceinline__ void tdm_load_1d(unsigned lds_byte_off, const void* gptr,
                                            unsigned tensor_len, unsigned tile_len) {
  u64 ga = (u64)(uintptr_t)gptr;
  v4u g0;
  g0.x = 1u;                                             // count=1, user descriptor
  g0.y = lds_byte_off;                                   // lds_addr
  g0.z = (u32)(ga & 0xFFFFFFFFu);                        // global_addr[31:0]
  g0.w = (u32)((ga >> 32) & 0x1FFFFFFu) | (2u << 30);    // global_addr[56:32] | type=2
  v8i g1;
  g1[0] = (int)(2u << 16);                               // data_size = 4 bytes
  g1[1] = (int)((tensor_len & 0xFFFFu) << 16);           // tensor_dim0[15:0]
  g1[2] = (int)(((tensor_len >> 16) & 0xFFFFu) | (1u << 16)); // dim0[31:16], tensor_dim1=1
  g1[3] = (int)((tile_len & 0xFFFFu) << 16);             // tensor_dim1[31:16]=0, tile_dim0
  g1[4] = 1;                                             // tile_dim1=1, tile_dim2 unused
  g1[5] = (int)tensor_len;                               // tensor_dim0_stride[31:0]
  g1[6] = 0;                                             // stride[47:32], dim1_stride lo
  g1[7] = 0;
  v4i gz4 = {0, 0, 0, 0};
  v8i gz8 = {0, 0, 0, 0, 0, 0, 0, 0};
  __builtin_amdgcn_tensor_load_to_lds(g0, g1, gz4, gz4, gz8, 0);
}

// ---------------------------------------------------------------------------
// Pass-0 fused: build sortable keys (clip(x,0,1) >= 0 => IEEE bits are
// unsigned-monotonic), copy payload, and produce the pass-0 8-bit histogram,
// digit-major: hist[d * nblocks + b].
// ---------------------------------------------------------------------------
__global__ void init_hist(const float* __restrict__ idxs, const float* __restrict__ arr,
                          u32* __restrict__ keys, float* __restrict__ vals,
                          u32* __restrict__ hist, int n, int nblocks) {
  __shared__ u32 h[256];
  const int t = threadIdx.x, b = blockIdx.x;
  h[t] = 0;
  __syncthreads();
  const int base = b * TILE + t * KPT;
  if (base + KPT <= n) {
    const float4* pi = (const float4*)(idxs + base);
    const float4* pa = (const float4*)(arr + base);
    uint4* pk = (uint4*)(keys + base);
    float4* pv = (float4*)(vals + base);
#pragma unroll
    for (int q = 0; q < KPT / 4; ++q) {
      float4 x = pi[q];
      uint4 k;
      k.x = __float_as_uint(fminf(fmaxf(x.x, 0.f), 1.f));
      k.y = __float_as_uint(fminf(fmaxf(x.y, 0.f), 1.f));
      k.z = __float_as_uint(fminf(fmaxf(x.z, 0.f), 1.f));
      k.w = __float_as_uint(fminf(fmaxf(x.w, 0.f), 1.f));
      pk[q] = k;
      pv[q] = pa[q];
      atomicAdd(&h[k.x & 255u], 1u);
      atomicAdd(&h[k.y & 255u], 1u);
      atomicAdd(&h[k.z & 255u], 1u);
      atomicAdd(&h[k.w & 255u], 1u);
    }
  } else {
    for (int e = 0; e < KPT; ++e) {
      int i = base + e;
      if (i < n) {
        u32 k = __float_as_uint(fminf(fmaxf(idxs[i], 0.f), 1.f));
        keys[i] = k;
        vals[i] = arr[i];
        atomicAdd(&h[k & 255u], 1u);
      }
    }
  }
  __syncthreads();
  hist[t * nblocks + b] = h[t];
}

// Per-block 8-bit histogram for passes 1..3 (reads permuted keys).
__global__ void hist8(const u32* __restrict__ keys, u32* __restrict__ hist,
                      int n, int nblocks, int shift) {
  __shared__ u32 h[256];
  const int t = threadIdx.x, b = blockIdx.x;
  h[t] = 0;
  __syncthreads();
  const int base = b * TILE + t * KPT;
  if (base + KPT <= n) {
    const uint4* p = (const uint4*)(keys + base);
#pragma unroll
    for (int q = 0; q < KPT / 4; ++q) {
      uint4 k = p[q];
      atomicAdd(&h[(k.x >> shift) & 255u], 1u);
      atomicAdd(&h[(k.y >> shift) & 255u], 1u);
      atomicAdd(&h[(k.z >> shift) & 255u], 1u);
      atomicAdd(&h[(k.w >> shift) & 255u], 1u);
    }
  } else {
    for (int e = 0; e < KPT; ++e) {
      int i = base + e;
      if (i < n) atomicAdd(&h[(keys[i] >> shift) & 255u], 1u);
    }
  }
  __syncthreads();
  hist[t * nblocks + b] = h[t];
}

// ---------------------------------------------------------------------------
// Hierarchical exclusive scan over hist (256*nblocks entries, ~1M):
//   scan_reduce: per-1024-chunk sums   scan_top: scan of chunk sums
//   scan_apply : in-place exclusive scan with chunk bases
// ---------------------------------------------------------------------------
__global__ void scan_reduce(const u32* __restrict__ h, u32* __restrict__ tmp, int total) {
  __shared__ u32 red[BLK];
  const int t = threadIdx.x, b = blockIdx.x;
  const int base = b * 1024 + t * 4;
  u32 s = 0;
  if (base + 4 <= total) {
    uint4 x = *(const uint4*)(h + base);
    s = x.x + x.y + x.z + x.w;
  } else {
    for (int i = 0; i < 4; ++i) if (base + i < total) s += h[base + i];
  }
  red[t] = s;
  __syncthreads();
  for (int off = BLK / 2; off > 0; off >>= 1) {
    if (t < off) red[t] += red[t + off];
    __syncthreads();
  }
  if (t == 0) tmp[b] = red[0];
}

__global__ void scan_top(u32* __restrict__ h, int total) {
  __shared__ u32 buf[1024];
  __shared__ u32 carry;
  const int t = threadIdx.x;
  if (t == 0) carry = 0;
  __syncthreads();
  for (int base = 0; base < total; base += 1024) {
    u32 v = (base + t < total) ? h[base + t] : 0u;
    buf[t] = v;
    __syncthreads();
#pragma unroll
    for (int off = 1; off < 1024; off <<= 1) {
      u32 x = (t >= off) ? buf[t - off] : 0u;
      __syncthreads();
      buf[t] += x;
      __syncthreads();
    }
    u32 excl = buf[t] - v + carry;
    if (base + t < total) h[base + t] = excl;
    u32 chunk_total = buf[1023];
    __syncthreads();
    if (t == 0) carry += chunk_total;
    __syncthreads();
  }
}

__global__ void scan_apply(u32* __restrict__ h, const u32* __restrict__ chunkExcl, int total) {
  __shared__ u32 buf[BLK];
  const int t = threadIdx.x, b = blockIdx.x;
  const int base = b * 1024 + t * 4;
  uint4 x = {0, 0, 0, 0};
  if (base + 4 <= total) x = *(const uint4*)(h + base);
  else {
    if (base + 0 < total) x.x = h[base + 0];
    if (base + 1 < total) x.y = h[base + 1];
    if (base + 2 < total) x.z = h[base + 2];
  }
  const u32 s = x.x + x.y + x.z + x.w;
  buf[t] = s;
  __syncthreads();
  for (int off = 1; off < BLK; off <<= 1) {
    u32 y = (t >= off) ? buf[t - off] : 0u;
    __syncthreads();
    buf[t] += y;
    __syncthreads();
  }
  u32 run = chunkExcl[b] + buf[t] - s;
  uint4 o;
  o.x = run; run += x.x;
  o.y = run; run += x.y;
  o.z = run; run += x.z;
  o.w = run;
  if (base + 4 <= total) *(uint4*)(h + base) = o;
  else {
    if (base + 0 < total) h[base + 0] = o.x;
    if (base + 1 < total) h[base + 1] = o.y;
    if (base + 2 < total) h[base + 2] = o.z;
  }
}

// ---------------------------------------------------------------------------
// 8-bit stable scatter. TDM stages the tile into LDS; two stable 4-bit
// counting sorts (lo nibble, then hi nibble) order the tile by the full 8-bit
// digit; final phase writes lane-consecutive tile positions to global
// (digit runs -> contiguous destinations -> coalesced stores).
// ---------------------------------------------------------------------------
__global__ void scatter8(const u32* __restrict__ keysIn, const float* __restrict__ valsIn,
                         u32* __restrict__ keysOut, float* __restrict__ valsOut,
                         const u32* __restrict__ hist, int n, int nblocks, int shift) {
  __shared__ u32 tileK[TILE];
  __shared__ float tileV[TILE];
  __shared__ u32 sc[16 * BLK];
  __shared__ u32 h256[256];
  __shared__ u32 ts256[256];
  __shared__ u32 gb[256];
  __shared__ u32 start16[16];

  const int t = threadIdx.x, b = blockIdx.x;
  const int tileBase = b * TILE;
  const int validTotal = min(TILE, n - tileBase);
  const int myBase = t * KPT;
  const int myValid = max(0, min(KPT, validTotal - myBase));

  // ---- async tensor load of the tile (wave 0 issues, EXEC-independent) ----
  if (t < 32) {
    tdm_load_1d(lds_off(&tileK[0]), keysIn + tileBase, (unsigned)(n - tileBase), TILE);
    tdm_load_1d(lds_off(&tileV[0]), valsIn + tileBase, (unsigned)(n - tileBase), TILE);
    __builtin_amdgcn_s_wait_tensorcnt(0);
  }
  h256[t] = 0;
#pragma unroll
  for (int d = 0; d < 16; ++d) sc[d * BLK + t] = 0;
  gb[t] = hist[t * nblocks + b];
  __syncthreads();

  // ---- read my chunk from LDS, count 256-bin + lo-nibble columns ----
  u32 k[KPT]; float v[KPT];
#pragma unroll
  for (int q = 0; q < KPT / 4; ++q) {
    uint4 kk = *(const uint4*)&tileK[myBase + 4 * q];
    float4 vv = *(const float4*)&tileV[myBase + 4 * q];
    k[4 * q + 0] = kk.x; k[4 * q + 1] = kk.y; k[4 * q + 2] = kk.z; k[4 * q + 3] = kk.w;
    v[4 * q + 0] = vv.x; v[4 * q + 1] = vv.y; v[4 * q + 2] = vv.z; v[4 * q + 3] = vv.w;
  }
#pragma unroll
  for (int e = 0; e < KPT; ++e) {
    if (e < myValid) {
      u32 d = (k[e] >> shift) & 255u;
      atomicAdd(&h256[d], 1u);
      sc[(d & 15u) * BLK + t] += 1;
    }
  }
  __syncthreads();

  // ---- tileStart prep: inclusive scan of h256 into ts256 ----
  ts256[t] = h256[t];
  __syncthreads();
  for (int off = 1; off < 256; off <<= 1) {
    u32 y = (t >= off) ? ts256[t - off] : 0u;
    __syncthreads();
    ts256[t] += y;
    __syncthreads();
  }
  // tileStart(d) = ts256[d] - h256[d]  (computed at use)

  // ================= pass A: stable sort by lo nibble =================
  u32 own[16];
#pragma unroll
  for (int d = 0; d < 16; ++d) own[d] = sc[d * BLK + t];
  for (int off = 1; off < BLK; off <<= 1) {
    u32 x[16];
#pragma unroll
    for (int d = 0; d < 16; ++d) x[d] = (t >= off) ? sc[d * BLK + t - off] : 0u;
    __syncthreads();
#pragma unroll
    for (int d = 0; d < 16; ++d) sc[d * BLK + t] += x[d];
    __syncthreads();
  }
  if (t < 16) {
    u32 s = 0;
    for (int l = 0; l < t; ++l) s += sc[l * BLK + (BLK - 1)];
    start16[t] = s;
  }
  __syncthreads();
#pragma unroll
  for (int d = 0; d < 16; ++d) sc[d * BLK + t] = start16[d] + sc[d * BLK + t] - own[d];
  __syncthreads();
#pragma unroll
  for (int e = 0; e < KPT; ++e) {
    if (e < myValid) {
      u32 l = (k[e] >> shift) & 15u;
      u32 pos = sc[l * BLK + t]++;
      tileK[pos] = k[e];
      tileV[pos] = v[e];
    }
  }
  __syncthreads();

  // ================= pass B: stable sort by hi nibble =================
  u32 k2[KPT]; float v2[KPT];
#pragma unroll
  for (int q = 0; q < KPT / 4; ++q) {
    uint4 kk = *(const uint4*)&tileK[myBase + 4 * q];
    float4 vv = *(const float4*)&tileV[myBase + 4 * q];
    k2[4 * q + 0] = kk.x; k2[4 * q + 1] = kk.y; k2[4 * q + 2] = kk.z; k2[4 * q + 3] = kk.w;
    v2[4 * q + 0] = vv.x; v2[4 * q + 1] = vv.y; v2[4 * q + 2] = vv.z; v2[4 * q + 3] = vv.w;
  }
#pragma unroll
  for (int d = 0; d < 16; ++d) sc[d * BLK + t] = 0;
  __syncthreads();
#pragma unroll
  for (int e = 0; e < KPT; ++e)
    if (myBase + e < validTotal) sc[(((k2[e] >> shift) >> 4) & 15u) * BLK + t] += 1;
  __syncthreads();
#pragma unroll
  for (int d = 0; d < 16; ++d) own[d] = sc[d * BLK + t];
  for (int off = 1; off < BLK; off <<= 1) {
    u32 x[16];
#pragma unroll
    for (int d = 0; d < 16; ++d) x[d] = (t >= off) ? sc[d * BLK + t - off] : 0u;
    __syncthreads();
#pragma unroll
    for (int d = 0; d < 16; ++d) sc[d * BLK + t] += x[d];
    __syncthreads();
  }
  if (t < 16) {
    u32 s = 0;
    for (int l = 0; l < t; ++l) s += sc[l * BLK + (BLK - 1)];
    start16[t] = s;
  }
  __syncthreads();
#pragma unroll
  for (int d = 0; d < 16; ++d) sc[d * BLK + t] = start16[d] + sc[d * BLK + t] - own[d];
  __syncthreads();
#pragma unroll
  for (int e = 0; e < KPT; ++e) {
    if (myBase + e < validTotal) {
      u32 hnib = ((k2[e] >> shift) >> 4) & 15u;
      u32 pos = sc[hnib * BLK + t]++;
      tileK[pos] = k2[e];
      tileV[pos] = v2[e];
    }
  }
  __syncthreads();

  // ---- final: lane-consecutive tile positions -> coalesced global scatter
#pragma unroll
  for (int e = 0; e < KPT; ++e) {
    int j = e * BLK + t;
    if (j < validTotal) {
      u32 kk = tileK[j];
      u32 d = (kk >> shift) & 255u;
      u32 dest = gb[d] + (u32)j - (ts256[d] - h256[d]);
      keysOut[dest] = kk;
      valsOut[dest] = tileV[j];
    }
  }
}

// ---------------------------------------------------------------------------
// Reduction stage 1: deterministic block partials of sum(relu(r[i]-r[i+1]))
// ---------------------------------------------------------------------------
__global__ void gap_partial(const float* __restrict__ r, float* __restrict__ part, int n) {
  __shared__ float red[BLK];
  const int t = threadIdx.x;
  float s = 0.0f;
  const int stride = gridDim.x * blockDim.x;
  for (int i = blockIdx.x * blockDim.x + t; i < n - 1; i += stride) {
    float g = r[i] - r[i + 1];
    s += fmaxf(g, 0.0f);
  }
  red[t] = s;
  __syncthreads();
  for (int off = BLK / 2; off > 0; off >>= 1) {
    if (t < off) red[t] += red[t + off];
    __syncthreads();
  }
  if (t == 0) part[blockIdx.x] = red[0];
}

// ---------------------------------------------------------------------------
// Reduction stage 2: one wave32, WMMA f32 16x16x4 with B = ones accumulates
// row sums of 64 partials/instruction; C's 16 identical columns => total =
// sum(C)/16 (exact). EXEC all-ones: uniform loop, 32-thread launch.
// ---------------------------------------------------------------------------
__global__ void gap_final(const float* __restrict__ part, float* __restrict__ out,
                          int m, float alpha) {
  const int lane = threadIdx.x;
  v8f c = {};
  v2f bones; bones.x = 1.0f; bones.y = 1.0f;
  for (int base = 0; base < m; base += 64) {
    v2f a;
    a.x = part[base + 2 * lane + 0];
    a.y = part[base + 2 * lane + 1];
    c = __builtin_amdgcn_wmma_f32_16x16x4_f32(false, a, false, bones,
                                              (short)0, c, false, false);
  }
  float s = c[0] + c[1] + c[2] + c[3] + c[4] + c[5] + c[6] + c[7];
  for (int off = 16; off > 0; off >>= 1) s += __shfl_down(s, off, 32);
  if (lane == 0) out[0] = (s * (1.0f / 16.0f)) * alpha;
}

// ---------------------------------------------------------------------------
extern "C" void kernel_launch(void* const* d_in, const int* in_sizes, int n_in,
                              void* d_out, int out_size, void* d_ws, size_t ws_size,
                              hipStream_t stream) {
  const float* arr = (const float*)d_in[0];   // "array"
  const float* idx = (const float*)d_in[1];   // "indices"
  float* out = (float*)d_out;
  const int n = in_sizes[0];
  const int nblocks = (n + TILE - 1) / TILE;
  const int total = 256 * nblocks;
  const int nch = (total + 1023) / 1024;
  const size_t np = (size_t)((n + 3) & ~3);

  u32*   keyA = (u32*)d_ws;
  u32*   keyB = keyA + np;
  float* valA = (float*)(keyB + np);
  float* valB = valA + np;
  u32*   hist = (u32*)(valB + np);
  u32*   ctmp = hist + (size_t)total;
  float* part = (float*)(ctmp + (size_t)nch);

  u32* ks = keyA; u32* kd = keyB;
  float* vs = valA; float* vd = valB;

  for (int p = 0; p < RPASSES; ++p) {
    const int shift = p * 8;
    if (p == 0)
      init_hist<<<nblocks, BLK, 0, stream>>>(idx, arr, keyA, valA, hist, n, nblocks);
    else
      hist8<<<nblocks, BLK, 0, stream>>>(ks, hist, n, nblocks, shift);
    scan_reduce<<<nch, BLK, 0, stream>>>(hist, ctmp, total);
    scan_top<<<1, 1024, 0, stream>>>(ctmp, nch);
    scan_apply<<<nch, BLK, 0, stream>>>(hist, ctmp, total);
    scatter8<<<nblocks, BLK, 0, stream>>>(ks, vs, kd, vd, hist, n, nblocks, shift);
    u32* tk = ks; ks = kd; kd = tk;
    float* tv = vs; vs = vd; vd = tv;
  }
  // p0: A->B, p1: B->A, p2: A->B, p3: B->A  => final sorted values in valA == vs

  gap_partial<<<RED_BLOCKS, BLK, 0, stream>>>(vs, part, n);
  gap_final<<<1, 32, 0, stream>>>(part, out, RED_BLOCKS, 0.1f);
}